// AttentionModel_66030827208879
// MI455X (gfx1250) — compile-verified
//
#include <hip/hip_runtime.h>

typedef float v2f __attribute__((ext_vector_type(2)));
typedef float v8f __attribute__((ext_vector_type(8)));

// exact param types for the gfx1250 async-load builtins (from hipcc diagnostics)
typedef int v4i_vs __attribute__((vector_size(16)));
typedef __attribute__((address_space(1))) v4i_vs* g4p;   // global int4*
typedef __attribute__((address_space(3))) v4i_vs* l4p;   // LDS int4*
typedef __attribute__((address_space(1))) int*     g1p;  // global int*
typedef __attribute__((address_space(3))) int*     l1p;  // LDS int*

#define BATCH 256
#define NSEQ  1001
#define DMODEL 128
#define KVSTRIDE 384   // NL * D
#define NTILES 63      // ceil(1001/16)

// ---------------------------------------------------------------------------
// gfx1250 async global->LDS helpers (ASYNCcnt-tracked), with asm fallback.
// ---------------------------------------------------------------------------
__device__ __forceinline__ void async_g2l_b128(void* lds, const void* g) {
#if __has_builtin(__builtin_amdgcn_global_load_async_to_lds_b128)
  __builtin_amdgcn_global_load_async_to_lds_b128((g4p)g, (l4p)lds, 0, 0);
#else
  unsigned loff = (unsigned)(uintptr_t)lds;   // flat LDS addr low bits = offset
  asm volatile("global_load_async_to_lds_b128 %0, %1, off"
               :: "v"(loff), "v"(g) : "memory");
#endif
}

__device__ __forceinline__ void async_g2l_b32(void* lds, const void* g) {
#if __has_builtin(__builtin_amdgcn_global_load_async_to_lds_b32)
  __builtin_amdgcn_global_load_async_to_lds_b32((g1p)g, (l1p)lds, 0, 0);
#else
  unsigned loff = (unsigned)(uintptr_t)lds;
  asm volatile("global_load_async_to_lds_b32 %0, %1, off"
               :: "v"(loff), "v"(g) : "memory");
#endif
}

__device__ __forceinline__ void wait_async0() {
#if __has_builtin(__builtin_amdgcn_s_wait_asynccnt)
  __builtin_amdgcn_s_wait_asynccnt(0);
#else
  asm volatile("s_wait_asynccnt 0x0" ::: "memory");
#endif
}

// ---------------------------------------------------------------------------
// Dense GEMM: out[256x128] = (relu?)(A[256x128] @ Wm[128x128] + bias) + res
// One wave (32 threads) per 16x16 output tile, fp32 WMMA 16x16x4.
// ---------------------------------------------------------------------------
__global__ __launch_bounds__(32)
void gemm128_kernel(const float* __restrict__ A, const float* __restrict__ Wm,
                    const float* __restrict__ bias, const float* __restrict__ res,
                    float* __restrict__ out, int relu) {
  const int m0 = blockIdx.x * 16;
  const int n0 = blockIdx.y * 16;
  const int lane = threadIdx.x;
  const int ll = lane & 15;
  const int lh = lane >> 4;            // 0: K {0,1}, 1: K {2,3}
  v8f c = {};
#pragma unroll 4
  for (int t = 0; t < 32; ++t) {
    const int k0 = t * 4 + lh * 2;
    v2f a, bf;
    a.x = A[(m0 + ll) * DMODEL + k0 + 0];
    a.y = A[(m0 + ll) * DMODEL + k0 + 1];
    bf.x = Wm[(k0 + 0) * DMODEL + n0 + ll];
    bf.y = Wm[(k0 + 1) * DMODEL + n0 + ll];
    c = __builtin_amdgcn_wmma_f32_16x16x4_f32(false, a, false, bf, (short)0, c,
                                              false, false);
  }
  const float bv = bias[n0 + ll];
  float vals[8];
#pragma unroll
  for (int v = 0; v < 8; ++v) vals[v] = c[v] + bv;
  if (relu) {
#pragma unroll
    for (int v = 0; v < 8; ++v) vals[v] = fmaxf(vals[v], 0.0f);
  }
  if (res) {
#pragma unroll
    for (int v = 0; v < 8; ++v)
      vals[v] += res[(m0 + v + lh * 8) * DMODEL + n0 + ll];
  }
#pragma unroll
  for (int v = 0; v < 8; ++v)
    out[(m0 + v + lh * 8) * DMODEL + n0 + ll] = vals[v];
}

// ---------------------------------------------------------------------------
// LayerNorm over last dim (128). One block (128 threads) per row.
// ---------------------------------------------------------------------------
__global__ __launch_bounds__(128)
void ln_kernel(const float* __restrict__ x, const float* __restrict__ g,
               const float* __restrict__ bb, float* __restrict__ out) {
  const int row = blockIdx.x;
  const int tid = threadIdx.x;
  __shared__ float red[4];
  const float v = x[row * DMODEL + tid];
  float s = v;
  for (int off = 16; off; off >>= 1) s += __shfl_xor(s, off, 32);
  if ((tid & 31) == 0) red[tid >> 5] = s;
  __syncthreads();
  const float mean = (red[0] + red[1] + red[2] + red[3]) * (1.0f / 128.0f);
  const float d = v - mean;
  float s2 = d * d;
  for (int off = 16; off; off >>= 1) s2 += __shfl_xor(s2, off, 32);
  __syncthreads();
  if ((tid & 31) == 0) red[tid >> 5] = s2;
  __syncthreads();
  const float var = (red[0] + red[1] + red[2] + red[3]) * (1.0f / 128.0f);
  out[row * DMODEL + tid] = d / sqrtf(var + 1e-5f) * g[tid] + bb[tid];
}

// ---------------------------------------------------------------------------
// Cross-attention (layers 0/1): one block (512 thr) per batch element.
// q + mask staged via async global->LDS; scores via WMMA (head-selective B
// columns); per-head softmax; fused V reduction.
// ---------------------------------------------------------------------------
__global__ __launch_bounds__(512)
void attn_kernel(const float* __restrict__ q, const float* __restrict__ Kp,
                 const float* __restrict__ Vp, const int* __restrict__ maskp,
                 float* __restrict__ at, int layer) {
  const int b = blockIdx.x;
  const int tid = threadIdx.x;
  __shared__ float s_lds[1008 * 8];
  __shared__ float q_lds[DMODEL];
  __shared__ float bias_lds[1008];
  __shared__ int   mask_lds[1008];

  // ---- async staging: q (32 x b128, 16B aligned) + mask (b32) ----
  if (tid < 32) async_g2l_b128(q_lds + tid * 4, q + b * DMODEL + tid * 4);
  for (int n = tid; n < NSEQ; n += 512)
    async_g2l_b32(mask_lds + n, maskp + (size_t)b * NSEQ + n);
  wait_async0();
  __syncthreads();
  for (int n = tid; n < 1008; n += 512)
    bias_lds[n] = (n < NSEQ && mask_lds[n]) ? -1.0e9f : 0.0f;
  __syncthreads();

  const int lane = tid & 31, wave = tid >> 5;   // 16 waves
  const int ll = lane & 15, lh = lane >> 4;
  const size_t base = (size_t)b * NSEQ * KVSTRIDE + (size_t)layer * DMODEL;

  // ---- scores: tiles of 16 rows, 8 head-columns per WMMA C tile ----
  for (int tile = wave; tile < NTILES; tile += 16) {
    const int n0 = tile * 16;
    int m = n0 + ll; if (m > NSEQ - 1) m = NSEQ - 1;   // clamp tail reads
    const float* Arow = Kp + base + (size_t)m * KVSTRIDE;
    v8f c = {};
#pragma unroll 4
    for (int t = 0; t < 32; ++t) {
      const int d0 = t * 4 + lh * 2;
      v2f a, bf;
      a.x = __builtin_nontemporal_load(Arow + d0);
      a.y = __builtin_nontemporal_load(Arow + d0 + 1);
      const int head = t >> 2;          // head owning K-dims [16h, 16h+16)
      const float q0 = q_lds[d0];
      const float q1 = q_lds[d0 + 1];
      bf.x = (ll == head) ? q0 : 0.0f;
      bf.y = (ll == head) ? q1 : 0.0f;
      c = __builtin_amdgcn_wmma_f32_16x16x4_f32(false, a, false, bf, (short)0,
                                                c, false, false);
    }
    if (ll < 8) {
#pragma unroll
      for (int v = 0; v < 8; ++v) {
        const int n = n0 + v + lh * 8;
        if (n < NSEQ) s_lds[n * 8 + ll] = c[v] * 0.25f;   // 1/sqrt(16)
      }
    }
  }
  __syncthreads();

  // ---- softmax: wave h handles head h (waves 8..15 idle here) ----
  if (wave < 8) {
    const int h = wave;
    float mx = -3.0e38f;
    for (int n = lane; n < NSEQ; n += 32)
      mx = fmaxf(mx, s_lds[n * 8 + h] + bias_lds[n]);
    for (int off = 16; off; off >>= 1) mx = fmaxf(mx, __shfl_xor(mx, off, 32));
    float sum = 0.0f;
    for (int n = lane; n < NSEQ; n += 32)
      sum += expf(s_lds[n * 8 + h] + bias_lds[n] - mx);
    for (int off = 16; off; off >>= 1) sum += __shfl_xor(sum, off, 32);
    const float inv = 1.0f / sum;
    for (int n = lane; n < NSEQ; n += 32)
      s_lds[n * 8 + h] = expf(s_lds[n * 8 + h] + bias_lds[n] - mx) * inv;
  }
  __syncthreads();

  // ---- output: out[d] = sum_n w[n, d/16] * V[b,n,d]  (4-way n split) ----
  const int d = tid & 127, grp = tid >> 7, hd = d >> 4;
  float acc = 0.0f;
  for (int n = grp; n < NSEQ; n += 4)
    acc += s_lds[n * 8 + hd] *
           __builtin_nontemporal_load(Vp + base + (size_t)n * KVSTRIDE + d);
  bias_lds[tid] = acc;                 // bias region free to reuse now
  __syncthreads();
  if (tid < DMODEL)
    at[b * DMODEL + tid] = bias_lds[tid] + bias_lds[tid + 128] +
                           bias_lds[tid + 256] + bias_lds[tid + 384];
}

// ---------------------------------------------------------------------------
// Final layer: single-head clipped-tanh attention -> softmax probs (output).
// ---------------------------------------------------------------------------
__global__ __launch_bounds__(512)
void final_attn_kernel(const float* __restrict__ q, const float* __restrict__ Kp,
                       const int* __restrict__ maskp, float* __restrict__ out) {
  const int b = blockIdx.x;
  const int tid = threadIdx.x;
  __shared__ float s_lds[1008];
  __shared__ float q_lds[DMODEL];
  __shared__ float bias_lds[1008];
  __shared__ int   mask_lds[1008];
  __shared__ float redA[16];
  __shared__ float redB[16];

  if (tid < 32) async_g2l_b128(q_lds + tid * 4, q + b * DMODEL + tid * 4);
  for (int n = tid; n < NSEQ; n += 512)
    async_g2l_b32(mask_lds + n, maskp + (size_t)b * NSEQ + n);
  wait_async0();
  __syncthreads();
  for (int n = tid; n < 1008; n += 512)
    bias_lds[n] = (n < NSEQ && mask_lds[n]) ? -1.0e9f : 0.0f;
  __syncthreads();

  const int lane = tid & 31, wave = tid >> 5;   // 16 waves
  const int ll = lane & 15, lh = lane >> 4;
  const size_t base = (size_t)b * NSEQ * KVSTRIDE + 2 * DMODEL;  // layer 2 K

  for (int tile = wave; tile < NTILES; tile += 16) {
    const int n0 = tile * 16;
    int m = n0 + ll; if (m > NSEQ - 1) m = NSEQ - 1;
    const float* Arow = Kp + base + (size_t)m * KVSTRIDE;
    v8f c = {};
#pragma unroll 4
    for (int t = 0; t < 32; ++t) {
      const int d0 = t * 4 + lh * 2;
      v2f a, bf;
      a.x = __builtin_nontemporal_load(Arow + d0);
      a.y = __builtin_nontemporal_load(Arow + d0 + 1);
      bf.x = q_lds[d0];                 // replicate q across all 16 columns
      bf.y = q_lds[d0 + 1];
      c = __builtin_amdgcn_wmma_f32_16x16x4_f32(false, a, false, bf, (short)0,
                                                c, false, false);
    }
    if (ll == 0) {                      // lanes 0 & 16 hold column 0
#pragma unroll
      for (int v = 0; v < 8; ++v) {
        const int n = n0 + v + lh * 8;
        if (n < NSEQ)
          s_lds[n] = 10.0f * tanhf(c[v] * 0.08838834764831845f); // 1/sqrt(128)
      }
    }
  }
  __syncthreads();

  float mx = -3.0e38f;
  for (int n = tid; n < NSEQ; n += 512)
    mx = fmaxf(mx, s_lds[n] + bias_lds[n]);
  for (int off = 16; off; off >>= 1) mx = fmaxf(mx, __shfl_xor(mx, off, 32));
  if (lane == 0) redA[wave] = mx;
  __syncthreads();
  mx = redA[0];
#pragma unroll
  for (int i = 1; i < 16; ++i) mx = fmaxf(mx, redA[i]);

  float sum = 0.0f;
  for (int n = tid; n < NSEQ; n += 512)
    sum += expf(s_lds[n] + bias_lds[n] - mx);
  for (int off = 16; off; off >>= 1) sum += __shfl_xor(sum, off, 32);
  if (lane == 0) redB[wave] = sum;
  __syncthreads();
  float tot = 0.0f;
#pragma unroll
  for (int i = 0; i < 16; ++i) tot += redB[i];
  const float inv = 1.0f / tot;
  for (int n = tid; n < NSEQ; n += 512)
    out[(size_t)b * NSEQ + n] = expf(s_lds[n] + bias_lds[n] - mx) * inv;
}

// ---------------------------------------------------------------------------
extern "C" void kernel_launch(void* const* d_in, const int* in_sizes, int n_in,
                              void* d_out, int out_size, void* d_ws, size_t ws_size,
                              hipStream_t stream) {
  (void)in_sizes; (void)n_in; (void)out_size; (void)ws_size;
  const float* h_t   = (const float*)d_in[0];
  const float* K_att = (const float*)d_in[1];
  const float* V_att = (const float*)d_in[2];
  const int*   mask  = (const int*)d_in[3];
  const float* Wt    = (const float*)d_in[4];   // (2, 8, 128, 128)
  const float* bt    = (const float*)d_in[5];   // (2, 8, 128)
  const float* ln_g  = (const float*)d_in[6];   // (2, 3, 128)
  const float* ln_b  = (const float*)d_in[7];
  const float* Wqf   = (const float*)d_in[8];
  const float* bqf   = (const float*)d_in[9];
  float* out = (float*)d_out;

  float* ws = (float*)d_ws;
  const int HD = BATCH * DMODEL;       // 32768 floats
  float* h  = ws;
  float* t1 = ws + HD;
  float* t2 = ws + 2 * HD;
  float* qb = ws + 3 * HD;
  float* at = ws + 4 * HD;

  const dim3 ggrid(BATCH / 16, DMODEL / 16);
  const dim3 gblk(32);

  for (int l = 0; l < 2; ++l) {
    const float* hin = (l == 0) ? h_t : h;
    const float* Wl  = Wt + (size_t)l * 8 * DMODEL * DMODEL;
    const float* bl  = bt + (size_t)l * 8 * DMODEL;
    const float* gl  = ln_g + (size_t)l * 3 * DMODEL;
    const float* bbl = ln_b + (size_t)l * 3 * DMODEL;

    // sa = v_sa = h @ W2 + b2   (S==1 self-attn collapses to V projection)
    gemm128_kernel<<<ggrid, gblk, 0, stream>>>(hin, Wl + 2 * 16384, bl + 2 * 128,
                                               nullptr, t1, 0);
    // t2 = h + sa @ W3 + b3 ; h = LN0(t2)
    gemm128_kernel<<<ggrid, gblk, 0, stream>>>(t1, Wl + 3 * 16384, bl + 3 * 128,
                                               hin, t2, 0);
    ln_kernel<<<BATCH, 128, 0, stream>>>(t2, gl + 0 * 128, bbl + 0 * 128, h);
    // q = h @ W4 + b4 ; at = CrossAttn(q, K_l, V_l, mask)
    gemm128_kernel<<<ggrid, gblk, 0, stream>>>(h, Wl + 4 * 16384, bl + 4 * 128,
                                               nullptr, qb, 0);
    attn_kernel<<<BATCH, 512, 0, stream>>>(qb, K_att, V_att, mask, at, l);
    // t2 = h + at @ W5 + b5 ; h = LN1(t2)
    gemm128_kernel<<<ggrid, gblk, 0, stream>>>(at, Wl + 5 * 16384, bl + 5 * 128,
                                               h, t2, 0);
    ln_kernel<<<BATCH, 128, 0, stream>>>(t2, gl + 1 * 128, bbl + 1 * 128, h);
    // t2 = h + relu(h @ W6 + b6) @ W7 + b7 ; h = LN2(t2)
    gemm128_kernel<<<ggrid, gblk, 0, stream>>>(h, Wl + 6 * 16384, bl + 6 * 128,
                                               nullptr, t1, 1);
    gemm128_kernel<<<ggrid, gblk, 0, stream>>>(t1, Wl + 7 * 16384, bl + 7 * 128,
                                               h, t2, 0);
    ln_kernel<<<BATCH, 128, 0, stream>>>(t2, gl + 2 * 128, bbl + 2 * 128, h);
  }

  // q_f = h @ Wq_final + bq_final ; probs = softmax(mask(10*tanh(qK/sqrt(128))))
  gemm128_kernel<<<ggrid, gblk, 0, stream>>>(h, Wqf, bqf, nullptr, qb, 0);
  final_attn_kernel<<<BATCH, 512, 0, stream>>>(qb, K_att, mask, out);
}